// CrossEntropyLoss_with_prob_20624432956049
// MI455X (gfx1250) — compile-verified
//
#include <hip/hip_runtime.h>

typedef float v2f __attribute__((ext_vector_type(2)));
typedef float v4f __attribute__((ext_vector_type(4)));
typedef float v8f __attribute__((ext_vector_type(8)));

#define CE_C 40
#define CE_VEC (CE_C / 4)
#define CE_BLOCKS 2048
#define CE_THREADS 256

// Wave32-wide sum using V_WMMA_F32_16X16X4_F32 against an all-ones B matrix.
// A layout (16x4 f32): lanes 0-15 VGPR0 = K=0 (our value), VGPR1 = K=1 (0);
//                      lanes 16-31 VGPR0 = K=2 (their value), VGPR1 = K=3 (0).
// => A row m = [v_m, 0, v_{m+16}, 0]; with B == 1 everywhere,
//    D[m,n] = v_m + v_{m+16} for every column n.
// D layout: lane n (0-15) holds column n, VGPR r = row r; lanes 16-31 hold
// rows r+8.  So sum of 8 D regs gives rows 0-7 (low half-lanes) / 8-15 (high
// half-lanes); one xor-16 shuffle completes the full 32-lane sum.
// Requires EXEC == all ones (callers keep every lane live).
__device__ __forceinline__ float wave_sum_wmma(float v) {
    v2f a; a[0] = v;    a[1] = 0.0f;
    v2f b; b[0] = 1.0f; b[1] = 1.0f;   // all-ones regardless of B layout
    v8f c = {};
    v8f d = __builtin_amdgcn_wmma_f32_16x16x4_f32(
        /*neg_a=*/false, a, /*neg_b=*/false, b,
        /*c_mod=*/(short)0, c, /*reuse_a=*/false, /*reuse_b=*/false);
    float s = d[0] + d[1] + d[2] + d[3] + d[4] + d[5] + d[6] + d[7];
    s += __shfl_xor(s, 16, 32);        // combine rows 0-7 with rows 8-15
    return s;                          // full wave sum in every lane
}

// Stage 1: per-block partial sums of per-row losses.
__global__ __launch_bounds__(CE_THREADS)
void ce_partial_kernel(const float* __restrict__ x, const float* __restrict__ t,
                       float* __restrict__ partial, int n) {
    __shared__ float wsum[CE_THREADS / 32];
    const int tid    = blockIdx.x * blockDim.x + threadIdx.x;
    const int stride = gridDim.x * blockDim.x;

    float local = 0.0f;
    for (int row = tid; row < n; row += stride) {
        const v4f* xr = (const v4f*)(x + (size_t)row * CE_C);
        const v4f* tr = (const v4f*)(t + (size_t)row * CE_C);
        v4f xv[CE_VEC], tv[CE_VEC];
        #pragma unroll
        for (int k = 0; k < CE_VEC; ++k) xv[k] = __builtin_nontemporal_load(xr + k);
        #pragma unroll
        for (int k = 0; k < CE_VEC; ++k) tv[k] = __builtin_nontemporal_load(tr + k);

        // row max
        float m = xv[0][0];
        #pragma unroll
        for (int k = 0; k < CE_VEC; ++k) {
            #pragma unroll
            for (int j = 0; j < 4; ++j) m = fmaxf(m, xv[k][j]);
        }
        // sum exp(x-m), dot(t,x), sum(t)
        float se = 0.0f, dot = 0.0f, st = 0.0f;
        #pragma unroll
        for (int k = 0; k < CE_VEC; ++k) {
            #pragma unroll
            for (int j = 0; j < 4; ++j) {
                se  += __expf(xv[k][j] - m);
                dot  = fmaf(tv[k][j], xv[k][j], dot);
                st  += tv[k][j];
            }
        }
        // loss = (m + log(se)) * sum(t) - dot(t, x)
        local += (m + __logf(se)) * st - dot;
    }

    // no divergent exits before here: EXEC is all ones for the WMMA
    float s = wave_sum_wmma(local);
    const int wid = threadIdx.x >> 5;
    if ((threadIdx.x & 31) == 0) wsum[wid] = s;
    __syncthreads();
    if (threadIdx.x == 0) {
        float bs = 0.0f;
        #pragma unroll
        for (int i = 0; i < CE_THREADS / 32; ++i) bs += wsum[i];
        partial[blockIdx.x] = bs;
    }
}

// Stage 2: one block reduces the block partials and writes mean.
__global__ __launch_bounds__(CE_THREADS)
void ce_final_kernel(const float* __restrict__ partial, float* __restrict__ out,
                     int nparts, float inv_n) {
    __shared__ float wsum[CE_THREADS / 32];
    float local = 0.0f;
    for (int i = threadIdx.x; i < nparts; i += blockDim.x) local += partial[i];

    float s = wave_sum_wmma(local);
    const int wid = threadIdx.x >> 5;
    if ((threadIdx.x & 31) == 0) wsum[wid] = s;
    __syncthreads();
    if (threadIdx.x == 0) {
        float tot = 0.0f;
        #pragma unroll
        for (int i = 0; i < CE_THREADS / 32; ++i) tot += wsum[i];
        out[0] = tot * inv_n;
    }
}

extern "C" void kernel_launch(void* const* d_in, const int* in_sizes, int n_in,
                              void* d_out, int out_size, void* d_ws, size_t ws_size,
                              hipStream_t stream) {
    const float* x = (const float*)d_in[0];   // input  [N, 40] f32
    const float* t = (const float*)d_in[1];   // target [N, 40] f32
    float* out     = (float*)d_out;           // scalar f32
    float* partial = (float*)d_ws;            // CE_BLOCKS floats of scratch

    const int n = in_sizes[0] / CE_C;         // number of rows

    ce_partial_kernel<<<CE_BLOCKS, CE_THREADS, 0, stream>>>(x, t, partial, n);
    ce_final_kernel<<<1, CE_THREADS, 0, stream>>>(partial, out, CE_BLOCKS,
                                                  1.0f / (float)n);
}